// Grapher_54022098649947
// MI455X (gfx1250) — compile-verified
//
#include <hip/hip_runtime.h>
#include <cmath>

typedef float v2f __attribute__((ext_vector_type(2)));
typedef float v8f __attribute__((ext_vector_type(8)));

#define BB 8
#define C1 96
#define C2 192
#define HH 160
#define WW 160
#define HW (HH*WW)
#define TPI (HH*(WW/16))      /* tiles per image = 1600 */
#define NTILES (BB*TPI)       /* 12800 */

/* ---- workspace layout (float offsets) ---- */
#define OFF_H   ((size_t)0)
#define SZ_H    ((size_t)BB*C1*HW)           /* 19,660,800 */
#define OFF_CM  (OFF_H + SZ_H)
#define SZ_CM   ((size_t)BB*C1*2*WW*2)       /* 491,520 */
#define OFF_RM  (OFF_CM + SZ_CM)
#define SZ_RM   ((size_t)BB*C1*HH*2*2)       /* 491,520 */
#define OFF_W1  (OFF_RM + SZ_RM)
#define SZ_W1   ((size_t)24*6*32*2)          /* 9,216 */
#define OFF_WG  (OFF_W1 + SZ_W1)
#define SZ_WG   ((size_t)48*12*32*2)         /* 36,864 */
#define OFF_W2  (OFF_WG + SZ_WG)
#define SZ_W2   ((size_t)48*6*32*2)          /* 18,432 */
#define OFF_B1  (OFF_W2 + SZ_W2)
#define OFF_BG  (OFF_B1 + 96)
#define OFF_B2  (OFF_BG + 192)

#define BIGF 3.0e38f

__device__ __forceinline__ v8f wmma4(v2f a, v2f b, v8f c) {
  /* D = A(16x4 f32) x B(4x16 f32) + C(16x16 f32) */
  return __builtin_amdgcn_wmma_f32_16x16x4_f32(false, a, false, b, (short)0, c,
                                               false, false);
}

/* ================= prep: fold BN into weights, swizzle to WMMA-B layout ===== */
__global__ void prep_kernel(
    const float* __restrict__ fc1_w, const float* __restrict__ fc1_b,
    const float* __restrict__ bn1_g, const float* __restrict__ bn1_b,
    const float* __restrict__ bn1_m, const float* __restrict__ bn1_v,
    const float* __restrict__ gc_w,  const float* __restrict__ gc_b,
    const float* __restrict__ bng_g, const float* __restrict__ bng_b,
    const float* __restrict__ bng_m, const float* __restrict__ bng_v,
    const float* __restrict__ fc2_w, const float* __restrict__ fc2_b,
    const float* __restrict__ bn2_g, const float* __restrict__ bn2_b,
    const float* __restrict__ bn2_m, const float* __restrict__ bn2_v,
    float* __restrict__ w1r, float* __restrict__ wgr, float* __restrict__ w2r,
    float* __restrict__ b1o, float* __restrict__ bgo, float* __restrict__ b2o)
{
  const int N_W1 = 24*6*32;          /* v2f entries */
  const int N_WG = 48*12*32;
  const int N_W2 = 48*6*32;
  int id = blockIdx.x * blockDim.x + threadIdx.x;
  if (id < N_W1) {
    int lane = id & 31, t = id >> 5, nt = t % 6, ks = t / 6;
    int o = nt*16 + (lane & 15);
    int k0 = ks*4 + (lane >> 4)*2;
    float s = bn1_g[o] / sqrtf(bn1_v[o] + 1e-5f);
    w1r[id*2+0] = fc1_w[o*C1 + k0+0] * s;
    w1r[id*2+1] = fc1_w[o*C1 + k0+1] * s;
  } else if (id < N_W1 + N_WG) {
    int j = id - N_W1;
    int lane = j & 31, t = j >> 5, nt = t % 12, ks = t / 12;
    int o = nt*16 + (lane & 15);
    int k0 = ks*4 + (lane >> 4)*2;
    float s = bng_g[o] / sqrtf(bng_v[o] + 1e-5f);
    wgr[j*2+0] = gc_w[o*C2 + k0+0] * s;
    wgr[j*2+1] = gc_w[o*C2 + k0+1] * s;
  } else if (id < N_W1 + N_WG + N_W2) {
    int j = id - N_W1 - N_WG;
    int lane = j & 31, t = j >> 5, nt = t % 6, ks = t / 6;
    int o = nt*16 + (lane & 15);
    int k0 = ks*4 + (lane >> 4)*2;
    float s = bn2_g[o] / sqrtf(bn2_v[o] + 1e-5f);
    w2r[j*2+0] = fc2_w[o*C2 + k0+0] * s;
    w2r[j*2+1] = fc2_w[o*C2 + k0+1] * s;
  } else {
    int j = id - N_W1 - N_WG - N_W2;
    if (j < 96) {
      float s = bn1_g[j] / sqrtf(bn1_v[j] + 1e-5f);
      b1o[j] = (fc1_b[j] - bn1_m[j]) * s + bn1_b[j];
    } else if (j < 96 + 192) {
      int o = j - 96;
      float s = bng_g[o] / sqrtf(bng_v[o] + 1e-5f);
      bgo[o] = (gc_b[o] - bng_m[o]) * s + bng_b[o];
    } else if (j < 96 + 192 + 96) {
      int o = j - 288;
      float s = bn2_g[o] / sqrtf(bn2_v[o] + 1e-5f);
      b2o[o] = (fc2_b[o] - bn2_m[o]) * s + bn2_b[o];
    }
  }
}

/* ================= fc1: h = BN(conv1x1(x)) via f32 WMMA ===================== */
__global__ __launch_bounds__(256) void fc1_kernel(
    const float* __restrict__ x, const float* __restrict__ w1r,
    const float* __restrict__ b1, float* __restrict__ h)
{
  int wave = threadIdx.x >> 5, lane = threadIdx.x & 31;
  int tile = blockIdx.x * 8 + wave;
  int b = tile / TPI, rem = tile % TPI;
  int y = rem / 10, x0 = (rem % 10) * 16;
  int m = lane & 15, kh = lane >> 4;

  const float* xp = x + (size_t)b*C1*HW + (size_t)y*WW + x0 + m;
  v8f zero = {0.f,0.f,0.f,0.f,0.f,0.f,0.f,0.f};
  v8f acc[6];
#pragma unroll
  for (int i = 0; i < 6; ++i) acc[i] = zero;

  for (int ks = 0; ks < 24; ++ks) {
    int k = ks*4 + kh*2;
    v2f a;
    a.x = xp[(size_t)k * HW];
    a.y = xp[(size_t)(k+1) * HW];
    const v2f* bw = (const v2f*)w1r + (size_t)(ks*6)*32 + lane;
#pragma unroll
    for (int nt = 0; nt < 6; ++nt)
      acc[nt] = wmma4(a, bw[nt*32], acc[nt]);
  }

  int n = lane & 15, mb = kh * 8;
#pragma unroll
  for (int nt = 0; nt < 6; ++nt) {
    int o = nt*16 + n;
    float bias = b1[o];
    float* hp = h + ((size_t)b*C1 + o)*HW + (size_t)y*WW + x0 + mb;
    float4 lo = { acc[nt][0]+bias, acc[nt][1]+bias, acc[nt][2]+bias, acc[nt][3]+bias };
    float4 hi = { acc[nt][4]+bias, acc[nt][5]+bias, acc[nt][6]+bias, acc[nt][7]+bias };
    *(float4*)hp = lo;
    *(float4*)(hp + 4) = hi;
  }
}

/* ============ mins: per-(b,c) parity min1/min2 tables over xs grid ========= */
__global__ void mins_kernel(const float* __restrict__ h,
                            float* __restrict__ cm, float* __restrict__ rm)
{
  int bc = blockIdx.x;            /* 0..767 */
  int t = threadIdx.x;            /* 0..159 */
  const float* plane = h + (size_t)bc * HW;

  /* column t of xs grid: min over rows per row-parity */
  {
    int wc = t + 1; if (wc >= WW) wc -= WW;
    float m1[2] = {BIGF, BIGF}, m2[2] = {BIGF, BIGF};
    for (int a = 0; a < HH; ++a) {
      int r = a + 1; if (r >= HH) r -= HH;
      float v = plane[r*WW + wc];
      int p = a & 1;
      if (v < m1[p]) { m2[p] = m1[p]; m1[p] = v; }
      else if (v < m2[p]) { m2[p] = v; }
    }
#pragma unroll
    for (int p = 0; p < 2; ++p) {
      size_t base = (((size_t)bc*2 + p)*WW + t)*2;
      cm[base+0] = m1[p]; cm[base+1] = m2[p];
    }
  }
  /* row t of xs grid: min over cols per col-parity */
  {
    int rr = t + 1; if (rr >= HH) rr -= HH;
    const float* rowp = plane + (size_t)rr*WW;
    float m1[2] = {BIGF, BIGF}, m2[2] = {BIGF, BIGF};
    for (int w = 0; w < WW; ++w) {
      int cidx = w + 1; if (cidx >= WW) cidx -= WW;
      float v = rowp[cidx];
      int p = w & 1;
      if (v < m1[p]) { m2[p] = m1[p]; m1[p] = v; }
      else if (v < m2[p]) { m2[p] = v; }
    }
#pragma unroll
    for (int p = 0; p < 2; ++p) {
      size_t base = (((size_t)bc*HH + t)*2 + p)*2;
      rm[base+0] = m1[p]; rm[base+1] = m2[p];
    }
  }
}

/* ====== fused: build g -> gc GEMM -> BN+GELU -> fc2 GEMM -> BN + residual === */
__global__ __launch_bounds__(128) void gc_fc2_kernel(
    const float* __restrict__ h,
    const float* __restrict__ cmt, const float* __restrict__ rmt,
    const float* __restrict__ wgr, const float* __restrict__ bg,
    const float* __restrict__ w2r, const float* __restrict__ b2,
    const float* __restrict__ xin, float* __restrict__ out)
{
  __shared__ float lds[4 * 3072];     /* 12KB per wave */
  int wave = threadIdx.x >> 5, lane = threadIdx.x & 31;
  float* gl = lds + wave * 3072;

  int tile = blockIdx.x * 4 + wave;
  int b = tile / TPI, rem = tile % TPI;
  int y = rem / 10, x0 = (rem % 10) * 16;
  int m = lane & 15, kh = lane >> 4;

  const float* hb = h + (size_t)b*C1*HW;

  /* --- build g lower half (k<96): g[k] = h[k] --- */
  {
    const float* hp = hb + (size_t)kh*HW + (size_t)y*WW + x0 + m;
    for (int cp = 0; cp < 48; ++cp)
      gl[cp*32 + m*2 + kh] = hp[(size_t)cp*2*HW];
  }
  /* --- build g upper half (k=96+c): xj via min-exclude tables --- */
  {
    int ys = y + 1; if (ys >= HH) ys -= HH;
    int xsx = x0 + m + 1; if (xsx >= WW) xsx -= WW;
    int pary = y & 1;
    int parw = (x0 + m) & 1;
    for (int cp = 0; cp < 48; ++cp) {
      int c = cp*2 + kh;
      float xsv = hb[(size_t)c*HW + (size_t)ys*WW + xsx];
      float2 cmv = *(const float2*)(cmt + ((((size_t)(b*C1 + c))*2 + pary)*WW + (x0+m))*2);
      float2 rmv = *(const float2*)(rmt + ((((size_t)(b*C1 + c))*HH + y)*2 + parw)*2);
      float cme = (xsv == cmv.x) ? cmv.y : cmv.x;
      float rme = (xsv == rmv.x) ? rmv.y : rmv.x;
      float xj = fmaxf(0.f, fmaxf(xsv - cme, xsv - rme));
      gl[(48 + cp)*32 + m*2 + kh] = xj;
    }
  }
  asm volatile("s_wait_dscnt 0x0" ::: "memory");

  /* --- GEMM1: [16 x 192] = g[16 x 192] * Wg^T, 48 K-steps x 12 N-tiles --- */
  v8f zero = {0.f,0.f,0.f,0.f,0.f,0.f,0.f,0.f};
  v8f acc1[12];
#pragma unroll
  for (int i = 0; i < 12; ++i) acc1[i] = zero;

  for (int ks = 0; ks < 48; ++ks) {
    v2f a = *(const v2f*)(gl + (ks*2 + kh)*32 + m*2);
    const v2f* bw = (const v2f*)wgr + (size_t)(ks*12)*32 + lane;
#pragma unroll
    for (int nt = 0; nt < 12; ++nt)
      acc1[nt] = wmma4(a, bw[nt*32], acc1[nt]);
  }

  /* --- bias + exact GELU, write g2 back into LDS in A-matrix layout --- */
  int n = lane & 15, mb = kh * 8;
#pragma unroll
  for (int nt = 0; nt < 12; ++nt) {
    int o = nt*16 + n;
    float bias = bg[o];
    int kp = o >> 1, kb = o & 1;
#pragma unroll
    for (int r = 0; r < 8; ++r) {
      float v = acc1[nt][r] + bias;
      v = 0.5f * v * (1.0f + erff(v * 0.70710678118654752f));
      gl[kp*32 + (mb + r)*2 + kb] = v;
    }
  }
  asm volatile("s_wait_dscnt 0x0" ::: "memory");

  /* --- GEMM2: [16 x 96] = g2[16 x 192] * W2^T, 48 K-steps x 6 N-tiles --- */
  v8f acc2[6];
#pragma unroll
  for (int i = 0; i < 6; ++i) acc2[i] = zero;

  for (int ks = 0; ks < 48; ++ks) {
    v2f a = *(const v2f*)(gl + (ks*2 + kh)*32 + m*2);
    const v2f* bw = (const v2f*)w2r + (size_t)(ks*6)*32 + lane;
#pragma unroll
    for (int nt = 0; nt < 6; ++nt)
      acc2[nt] = wmma4(a, bw[nt*32], acc2[nt]);
  }

  /* --- bias + residual + store --- */
#pragma unroll
  for (int nt = 0; nt < 6; ++nt) {
    int o = nt*16 + n;
    float bias = b2[o];
    size_t off = ((size_t)b*C1 + o)*HW + (size_t)y*WW + x0 + mb;
    const float4 rlo = *(const float4*)(xin + off);
    const float4 rhi = *(const float4*)(xin + off + 4);
    float4 lo = { acc2[nt][0]+bias+rlo.x, acc2[nt][1]+bias+rlo.y,
                  acc2[nt][2]+bias+rlo.z, acc2[nt][3]+bias+rlo.w };
    float4 hi = { acc2[nt][4]+bias+rhi.x, acc2[nt][5]+bias+rhi.y,
                  acc2[nt][6]+bias+rhi.z, acc2[nt][7]+bias+rhi.w };
    *(float4*)(out + off) = lo;
    *(float4*)(out + off + 4) = hi;
  }
}

/* ========================================================================== */
extern "C" void kernel_launch(void* const* d_in, const int* in_sizes, int n_in,
                              void* d_out, int out_size, void* d_ws, size_t ws_size,
                              hipStream_t stream) {
  const float* x     = (const float*)d_in[0];
  const float* fc1_w = (const float*)d_in[1];
  const float* fc1_b = (const float*)d_in[2];
  const float* bn1_g = (const float*)d_in[3];
  const float* bn1_b = (const float*)d_in[4];
  const float* bn1_m = (const float*)d_in[5];
  const float* bn1_v = (const float*)d_in[6];
  const float* gc_w  = (const float*)d_in[7];
  const float* gc_b  = (const float*)d_in[8];
  const float* bng_g = (const float*)d_in[9];
  const float* bng_b = (const float*)d_in[10];
  const float* bng_m = (const float*)d_in[11];
  const float* bng_v = (const float*)d_in[12];
  const float* fc2_w = (const float*)d_in[13];
  const float* fc2_b = (const float*)d_in[14];
  const float* bn2_g = (const float*)d_in[15];
  const float* bn2_b = (const float*)d_in[16];
  const float* bn2_m = (const float*)d_in[17];
  const float* bn2_v = (const float*)d_in[18];

  float* ws  = (float*)d_ws;
  float* h   = ws + OFF_H;
  float* cm  = ws + OFF_CM;
  float* rm  = ws + OFF_RM;
  float* w1r = ws + OFF_W1;
  float* wgr = ws + OFF_WG;
  float* w2r = ws + OFF_W2;
  float* b1o = ws + OFF_B1;
  float* bgo = ws + OFF_BG;
  float* b2o = ws + OFF_B2;

  prep_kernel<<<128, 256, 0, stream>>>(
      fc1_w, fc1_b, bn1_g, bn1_b, bn1_m, bn1_v,
      gc_w,  gc_b,  bng_g, bng_b, bng_m, bng_v,
      fc2_w, fc2_b, bn2_g, bn2_b, bn2_m, bn2_v,
      w1r, wgr, w2r, b1o, bgo, b2o);

  fc1_kernel<<<NTILES/8, 256, 0, stream>>>(x, w1r, b1o, h);

  mins_kernel<<<BB*C1, 160, 0, stream>>>(h, cm, rm);

  gc_fc2_kernel<<<NTILES/4, 128, 0, stream>>>(h, cm, rm, wgr, bgo, w2r, b2o,
                                              x, (float*)d_out);
}